// Multi_SAE_8546984919259
// MI455X (gfx1250) — compile-verified
//
#include <hip/hip_runtime.h>
#include <hip/hip_bf16.h>
#include <math.h>

#define NB 2048          // batch
#define NTOK 3

typedef __attribute__((ext_vector_type(2))) float v2f;
typedef __attribute__((ext_vector_type(8))) float v8f;

// ---------------------------------------------------------------------------
// e11 = tanh(X1 @ W(64,16) + b) via V_WMMA_F32_16X16X4_F32.
// One wave per 16x16 output tile, K=64 in 16 steps of 4.
// A 16x4 layout: lanes 0-15 M=0..15; VGPR0 holds K={0|2}, VGPR1 K={1|3} per half.
// C/D: VGPR r -> row r (lanes 0-15) / row 8+r (lanes 16-31), col = lane&15.
// ---------------------------------------------------------------------------
__global__ __launch_bounds__(32) void e11_wmma_kernel(const float* __restrict__ X1,
                                                      const float* __restrict__ W,
                                                      const float* __restrict__ bias,
                                                      float* __restrict__ out) {
  const int m0   = blockIdx.x * 16;
  const int lane = threadIdx.x & 31;
  const int half = lane >> 4;
  const int l    = lane & 15;
  v8f c = {};
#pragma unroll
  for (int k = 0; k < 64; k += 4) {
    const int ka = k + half * 2;
    v2f a, bf;
    a.x  = X1[(m0 + l) * 64 + ka + 0];
    a.y  = X1[(m0 + l) * 64 + ka + 1];
    bf.x = W[(ka + 0) * 16 + l];
    bf.y = W[(ka + 1) * 16 + l];
    c = __builtin_amdgcn_wmma_f32_16x16x4_f32(false, a, false, bf, (short)0, c,
                                              false, false);
  }
#pragma unroll
  for (int r = 0; r < 8; ++r) {
    const int row = m0 + half * 8 + r;
    out[row * 16 + l] = tanhf(c[r] + bias[l]);
  }
}

// ---------------------------------------------------------------------------
// Output offsets (floats): x_recon(2048,136), h1(,6), h2(,4), h3(,3),
// xr1(,64), xr2(,40), xr3(,32), concatenated flat in return order.
// ---------------------------------------------------------------------------
#define OFF_XREC 0
#define OFF_H1 278528
#define OFF_H2 290816
#define OFF_H3 299008
#define OFF_XR1 305152
#define OFF_XR2 436224
#define OFF_XR3 518144

// ---------------------------------------------------------------------------
// xr1 = t2 @ d13_W(16,64) + b via WMMA; fused dual write (x_recon + xr1).
// grid.x = (2048/16) * 4 tiles; K=16 in 4 steps of 4.
// ---------------------------------------------------------------------------
__global__ __launch_bounds__(32) void d13_wmma_kernel(const float* __restrict__ t2buf,
                                                      const float* __restrict__ W,
                                                      const float* __restrict__ bias,
                                                      float* __restrict__ dout) {
  const int m0   = (blockIdx.x >> 2) * 16;
  const int n0   = (blockIdx.x & 3) * 16;
  const int lane = threadIdx.x & 31;
  const int half = lane >> 4;
  const int l    = lane & 15;
  v8f c = {};
#pragma unroll
  for (int k = 0; k < 16; k += 4) {
    const int ka = k + half * 2;
    v2f a, bf;
    a.x  = t2buf[(m0 + l) * 16 + ka + 0];
    a.y  = t2buf[(m0 + l) * 16 + ka + 1];
    bf.x = W[(ka + 0) * 64 + n0 + l];
    bf.y = W[(ka + 1) * 64 + n0 + l];
    c = __builtin_amdgcn_wmma_f32_16x16x4_f32(false, a, false, bf, (short)0, c,
                                              false, false);
  }
#pragma unroll
  for (int r = 0; r < 8; ++r) {
    const int row = m0 + half * 8 + r;
    const int col = n0 + l;
    const float v = c[r] + bias[col];
    dout[OFF_XREC + (size_t)row * 136 + col] = v;
    dout[OFF_XR1 + (size_t)row * 64 + col]   = v;
  }
}

// ---------------------------------------------------------------------------
// shared helpers
// ---------------------------------------------------------------------------
__device__ __forceinline__ void row_stats10(const float* r, float& Lse, float& H,
                                            float& A) {
  float m = r[0];
#pragma unroll
  for (int c = 1; c < 10; ++c) m = fmaxf(m, r[c]);
  float s = 0.f;
#pragma unroll
  for (int c = 0; c < 10; ++c) s += __expf(r[c] - m);
  Lse = m + __logf(s);
  float h = 0.f, a = 0.f;
#pragma unroll
  for (int c = 0; c < 10; ++c) {
    float p = __expf(r[c] - Lse);
    h += p * (r[c] - Lse);   // mean(softmax * log_softmax) numerator
    a += p * r[c];
  }
  H = h * 0.1f;
  A = a;
}

#define MATVEC(outv, n_out, inv, n_in, Wp, bp)                                 \
  _Pragma("unroll") for (int c_ = 0; c_ < (n_out); ++c_) {                     \
    float o_ = (bp)[c_];                                                       \
    _Pragma("unroll") for (int r_ = 0; r_ < (n_in); ++r_) o_ +=                \
        (inv)[r_] * (Wp)[r_ * (n_out) + c_];                                   \
    (outv)[c_] = o_;                                                           \
  }
#define MATVEC_TANH(outv, n_out, inv, n_in, Wp, bp)                            \
  _Pragma("unroll") for (int c_ = 0; c_ < (n_out); ++c_) {                     \
    float o_ = (bp)[c_];                                                       \
    _Pragma("unroll") for (int r_ = 0; r_ < (n_in); ++r_) o_ +=                \
        (inv)[r_] * (Wp)[r_ * (n_out) + c_];                                   \
    (outv)[c_] = tanhf(o_);                                                    \
  }

// ---------------------------------------------------------------------------
// Per-sample encoder tail: e12/h1, e21/e22/h2, e31/e32/h3, B1/B2/B3 -> Qb
// ---------------------------------------------------------------------------
__global__ void encoder_rest_kernel(
    const float* __restrict__ X2, const float* __restrict__ X3,
    const float* __restrict__ e11buf, const float* e12W, const float* e12b,
    const float* a1W, const float* a1b, const float* e21W, const float* e21b,
    const float* e22W, const float* e22b, const float* a2W, const float* a2b,
    const float* e31W, const float* e31b, const float* e32W, const float* e32b,
    const float* a3W, const float* a3b, const float* B1W, const float* B1b,
    const float* B2W, const float* B2b, const float* B3W, const float* B3b,
    float* __restrict__ Qb, float* __restrict__ h1o, float* __restrict__ h2o,
    float* __restrict__ h3o) {
  int i = blockIdx.x * blockDim.x + threadIdx.x;
  if (i >= NB) return;
  float e11[16], e12[11], h1[6], e21[10], e22[6], h2[4], e31[9], e32[5], h3[3];
#pragma unroll
  for (int r = 0; r < 16; ++r) e11[r] = e11buf[i * 16 + r];
  MATVEC_TANH(e12, 11, e11, 16, e12W, e12b);
  MATVEC(h1, 6, e12, 11, a1W, a1b);
  MATVEC_TANH(e21, 10, (X2 + i * 40), 40, e21W, e21b);
  MATVEC_TANH(e22, 6, e21, 10, e22W, e22b);
  MATVEC(h2, 4, e22, 6, a2W, a2b);
  MATVEC_TANH(e31, 9, (X3 + i * 32), 32, e31W, e31b);
  MATVEC_TANH(e32, 5, e31, 9, e32W, e32b);
  MATVEC(h3, 3, e32, 5, a3W, a3b);
#pragma unroll
  for (int r = 0; r < 6; ++r) h1o[i * 6 + r] = h1[r];
#pragma unroll
  for (int r = 0; r < 4; ++r) h2o[i * 4 + r] = h2[r];
#pragma unroll
  for (int r = 0; r < 3; ++r) h3o[i * 3 + r] = h3[r];
#pragma unroll
  for (int c = 0; c < 10; ++c) {  // f1 = concat(e11,e12,h1)(33) @ B1
    float o = B1b[c];
#pragma unroll
    for (int r = 0; r < 16; ++r) o += e11[r] * B1W[r * 10 + c];
#pragma unroll
    for (int r = 0; r < 11; ++r) o += e12[r] * B1W[(16 + r) * 10 + c];
#pragma unroll
    for (int r = 0; r < 6; ++r) o += h1[r] * B1W[(27 + r) * 10 + c];
    Qb[i * 30 + 0 + c] = o;
  }
#pragma unroll
  for (int c = 0; c < 10; ++c) {
    float o = B2b[c];
#pragma unroll
    for (int r = 0; r < 10; ++r) o += e21[r] * B2W[r * 10 + c];
#pragma unroll
    for (int r = 0; r < 6; ++r) o += e22[r] * B2W[(10 + r) * 10 + c];
#pragma unroll
    for (int r = 0; r < 4; ++r) o += h2[r] * B2W[(16 + r) * 10 + c];
    Qb[i * 30 + 10 + c] = o;
  }
#pragma unroll
  for (int c = 0; c < 10; ++c) {
    float o = B3b[c];
#pragma unroll
    for (int r = 0; r < 9; ++r) o += e31[r] * B3W[r * 10 + c];
#pragma unroll
    for (int r = 0; r < 5; ++r) o += e32[r] * B3W[(9 + r) * 10 + c];
#pragma unroll
    for (int r = 0; r < 3; ++r) o += h3[r] * B3W[(14 + r) * 10 + c];
    Qb[i * 30 + 20 + c] = o;
  }
}

// ---------------------------------------------------------------------------
// 3-token MI attention (heads 0 and 1), one thread per sample. Includes the
// running-sum-over-queries "bug" and the final Wf projection.
// ---------------------------------------------------------------------------
__global__ void attn3_kernel(const float* __restrict__ X, const float* Wq,
                             const float* bq, const float* Wk, const float* bk,
                             const float* Wv, const float* bv, const float* Wf,
                             const float* bf, float* __restrict__ out) {
  int i = blockIdx.x * blockDim.x + threadIdx.x;
  if (i >= NB) return;
  const float* x = X + i * 30;
  float Q[NTOK][10], K[NTOK][10], V[NTOK][10];
#pragma unroll
  for (int t = 0; t < NTOK; ++t)
#pragma unroll
    for (int c = 0; c < 10; ++c) {
      float aq = bq[c], ak = bk[c], av = bv[c];
#pragma unroll
      for (int r = 0; r < 10; ++r) {
        float xv = x[t * 10 + r];
        aq += xv * Wq[r * 10 + c];
        ak += xv * Wk[r * 10 + c];
        av += xv * Wv[r * 10 + c];
      }
      Q[t][c] = aq; K[t][c] = ak; V[t][c] = av;
    }
  float Lq[NTOK], HQ[NTOK], Aq[NTOK], Lk[NTOK], HK[NTOK], Ak[NTOK];
#pragma unroll
  for (int t = 0; t < NTOK; ++t) {
    row_stats10(Q[t], Lq[t], HQ[t], Aq[t]);
    row_stats10(K[t], Lk[t], HK[t], Ak[t]);
  }
  const float inv2d = 0.05f;                    // 1/(2*10)
  const float invScale = 0.44721359549995793f;  // 1/sqrt(10//2)
  float y[NTOK][10];
#pragma unroll
  for (int t = 0; t < NTOK; ++t) {
    float I[NTOK];
#pragma unroll
    for (int u = 0; u < NTOK; ++u) {
      float aa = Lq[t], bb = Lk[u];
      float mx = fmaxf(aa, bb), mn = fminf(aa, bb);
      float L = mx + log1pf(__expf(mn - mx));   // logaddexp
      float wq = __expf(aa - L), wk = __expf(bb - L);
      float HQK = (wq * (Aq[t] - L) + wk * (Ak[u] - L)) * inv2d;
      I[u] = -(HQ[t] + HK[u] - HQK) * invScale;
    }
    float m = fmaxf(I[0], fmaxf(I[1], I[2]));
    float e0 = __expf(I[0] - m), e1 = __expf(I[1] - m), e2 = __expf(I[2] - m);
    float inv = 1.f / (e0 + e1 + e2);
    e0 *= inv; e1 *= inv; e2 *= inv;
#pragma unroll
    for (int c = 0; c < 10; ++c)
      y[t][c] = e0 * V[0][c] + e1 * V[1][c] + e2 * V[2][c];
  }
#pragma unroll
  for (int t = 1; t < NTOK; ++t)
#pragma unroll
    for (int c = 0; c < 10; ++c) y[t][c] += y[t - 1][c];  // cumsum over tokens
#pragma unroll
  for (int t = 0; t < NTOK; ++t)
#pragma unroll
    for (int c = 0; c < 10; ++c) {
      float o = bf[c];
#pragma unroll
      for (int r = 0; r < 10; ++r) o += y[t][r] * Wf[r * 10 + c];
      out[i * 30 + t * 10 + c] = o;
    }
}

// out = concat(A, Qb)(20) @ L1_W + b, per (sample,token)
__global__ void lin_l1_kernel(const float* __restrict__ A,
                              const float* __restrict__ Qb, const float* W,
                              const float* b, float* __restrict__ out) {
  int idx = blockIdx.x * blockDim.x + threadIdx.x;
  if (idx >= NB * NTOK) return;
  const float* a = A + idx * 10;
  const float* q = Qb + idx * 10;
#pragma unroll
  for (int c = 0; c < 10; ++c) {
    float o = b[c];
#pragma unroll
    for (int r = 0; r < 10; ++r)
      o += a[r] * W[r * 10 + c] + q[r] * W[(10 + r) * 10 + c];
    out[idx * 10 + c] = o;
  }
}

// ---------------------------------------------------------------------------
// Head-2 attention over the 2048-long sample axis (3 sequences).
// Stage 1: per-(s,i) Q/K stats + V row.
// ---------------------------------------------------------------------------
__global__ void seqattn_stats_kernel(const float* __restrict__ Xs, /*(B,3,10)*/
                                     const float* Wq, const float* bq,
                                     const float* Wk, const float* bk,
                                     const float* Wv, const float* bv,
                                     float* __restrict__ Lq, float* __restrict__ HQ,
                                     float* __restrict__ Aq, float* __restrict__ Lk,
                                     float* __restrict__ HK, float* __restrict__ Ak,
                                     float* __restrict__ Vbuf) {
  int g = blockIdx.x * blockDim.x + threadIdx.x;
  if (g >= NTOK * NB) return;
  int s = g >> 11, i = g & (NB - 1);
  const float* x = Xs + (i * NTOK + s) * 10;
  float Q[10], K[10];
#pragma unroll
  for (int c = 0; c < 10; ++c) {
    float aq = bq[c], ak = bk[c], av = bv[c];
#pragma unroll
    for (int r = 0; r < 10; ++r) {
      aq += x[r] * Wq[r * 10 + c];
      ak += x[r] * Wk[r * 10 + c];
      av += x[r] * Wv[r * 10 + c];
    }
    Q[c] = aq; K[c] = ak;
    Vbuf[g * 10 + c] = av;
  }
  row_stats10(Q, Lq[g], HQ[g], Aq[g]);
  row_stats10(K, Lk[g], HK[g], Ak[g]);
}

// ---------------------------------------------------------------------------
// Stage 2: 8 waves/WG = 8 query rows (all in same sequence s since 8|2048).
// V[s] (80KB) + Lk/HK/Ak[s] (24KB) staged once into dynamic LDS and reused by
// all 8 rows (~8x cut in L2 traffic). Online softmax per lane, wave32
// __shfl_xor merge of (max, sum, acc[10]).
// ---------------------------------------------------------------------------
__global__ __launch_bounds__(256) void seqattn_rows_kernel(
    const float* __restrict__ Lq, const float* __restrict__ HQ,
    const float* __restrict__ Aq, const float* __restrict__ Lk,
    const float* __restrict__ HK, const float* __restrict__ Ak,
    const float* __restrict__ Vbuf, float* __restrict__ yp) {
  extern __shared__ float smem[];
  float* Vl  = smem;                 // 2048*10
  float* Lkl = smem + NB * 10;       // 2048
  float* Hkl = Lkl + NB;             // 2048
  float* Akl = Hkl + NB;             // 2048
  const int w = threadIdx.x >> 5;
  const int lane = threadIdx.x & 31;
  const int g = blockIdx.x * 8 + w;  // row id in [0, 3*2048)
  const int s = g >> 11;
  // cooperative staging (vectorized b128 both sides)
  {
    const float4* Vg = (const float4*)(Vbuf + (size_t)s * NB * 10);
    float4* Vd = (float4*)Vl;
    for (int t = threadIdx.x; t < NB * 10 / 4; t += 256) Vd[t] = Vg[t];
    const float4* Lg = (const float4*)(Lk + s * NB);
    const float4* Hg = (const float4*)(HK + s * NB);
    const float4* Ag = (const float4*)(Ak + s * NB);
    float4* Ld = (float4*)Lkl;
    float4* Hd = (float4*)Hkl;
    float4* Ad = (float4*)Akl;
    for (int t = threadIdx.x; t < NB / 4; t += 256) {
      Ld[t] = Lg[t]; Hd[t] = Hg[t]; Ad[t] = Ag[t];
    }
  }
  __syncthreads();
  const float Lqi = Lq[g], HQi = HQ[g], Aqi = Aq[g];
  float m = -3.4e38f, ssum = 0.f, acc[10];
#pragma unroll
  for (int c = 0; c < 10; ++c) acc[c] = 0.f;
  for (int j = lane; j < NB; j += 32) {
    float bb = Lkl[j];
    float mx = fmaxf(Lqi, bb), mn = fminf(Lqi, bb);
    float L = mx + log1pf(__expf(mn - mx));
    float wq = __expf(Lqi - L), wk = __expf(bb - L);
    float HQK = (wq * (Aqi - L) + wk * (Akl[j] - L)) * 0.05f;
    float I = -(HQi + Hkl[j] - HQK) * 0.44721359549995793f;
    const float* vj = Vl + j * 10;
    if (I <= m) {
      float p = __expf(I - m);
      ssum += p;
#pragma unroll
      for (int c = 0; c < 10; ++c) acc[c] += p * vj[c];
    } else {
      float r = __expf(m - I);
      ssum = ssum * r + 1.f;
#pragma unroll
      for (int c = 0; c < 10; ++c) acc[c] = acc[c] * r + vj[c];
      m = I;
    }
  }
  // wave32 merge of online-softmax partials
#pragma unroll
  for (int off = 16; off > 0; off >>= 1) {
    float mo = __shfl_xor(m, off, 32);
    float so = __shfl_xor(ssum, off, 32);
    float vo[10];
#pragma unroll
    for (int c = 0; c < 10; ++c) vo[c] = __shfl_xor(acc[c], off, 32);
    float M = fmaxf(m, mo);
    float s1 = __expf(m - M), s2 = __expf(mo - M);
    ssum = ssum * s1 + so * s2;
#pragma unroll
    for (int c = 0; c < 10; ++c) acc[c] = acc[c] * s1 + vo[c] * s2;
    m = M;
  }
  if (lane == 0) {
    float inv = 1.f / ssum;
#pragma unroll
    for (int c = 0; c < 10; ++c) yp[(size_t)g * 10 + c] = acc[c] * inv;
  }
}

// Stage 3: sequential cumsum over the 2048 sample axis, one lane per (s,col).
__global__ void seq_cumsum_kernel(float* __restrict__ yp) {
  int t = threadIdx.x;
  if (t >= 30) return;
  int s = t / 10, c = t % 10;
  float run = 0.f;
  float* base = yp + (size_t)s * NB * 10 + c;
  for (int i = 0; i < NB; ++i) { run += base[i * 10]; base[i * 10] = run; }
}

// Stage 4: project by Wf and transpose back to (B,3,10).
__global__ void seqattn_out_kernel(const float* __restrict__ yp, const float* Wf,
                                   const float* bf, float* __restrict__ attn_sample) {
  int g = blockIdx.x * blockDim.x + threadIdx.x;
  if (g >= NTOK * NB) return;
  int s = g >> 11, i = g & (NB - 1);
  const float* yrow = yp + (size_t)g * 10;
#pragma unroll
  for (int c = 0; c < 10; ++c) {
    float o = bf[c];
#pragma unroll
    for (int r = 0; r < 10; ++r) o += yrow[r] * Wf[r * 10 + c];
    attn_sample[(i * NTOK + s) * 10 + c] = o;
  }
}

// ---------------------------------------------------------------------------
// Decoders. Modality 1 stops at t2 (stored to ws) -> d13 handled by WMMA.
// Modalities 2/3 (N=40/32, K=10/9: not tile-friendly) finish in VALU.
// ---------------------------------------------------------------------------
__global__ void decoder_kernel(
    const float* __restrict__ Qb, const float* __restrict__ sup_attn,
    const float* __restrict__ attn_sample, const float* d11W, const float* d11b,
    const float* d12W, const float* d12b, const float* d21W, const float* d21b,
    const float* d22W, const float* d22b, const float* d23W, const float* d23b,
    const float* d31W, const float* d31b, const float* d32W, const float* d32b,
    const float* d33W, const float* d33b, float* __restrict__ t2buf,
    float* __restrict__ dout) {
  int i = blockIdx.x * blockDim.x + threadIdx.x;
  if (i >= NB) return;
  float o[30];
  // modality 1 -> t2 only
#pragma unroll
  for (int r = 0; r < 10; ++r) {
    o[r]      = Qb[(i * 3 + 0) * 10 + r];
    o[10 + r] = sup_attn[(i * 3 + 0) * 10 + r];
    o[20 + r] = attn_sample[(i * 3 + 0) * 10 + r];
  }
  {
    float t1[11], t2[16];
    MATVEC_TANH(t1, 11, o, 30, d11W, d11b);
    MATVEC_TANH(t2, 16, t1, 11, d12W, d12b);
#pragma unroll
    for (int c = 0; c < 16; ++c) t2buf[i * 16 + c] = t2[c];
  }
  // modality 2
#pragma unroll
  for (int r = 0; r < 10; ++r) {
    o[r]      = Qb[(i * 3 + 1) * 10 + r];
    o[10 + r] = sup_attn[(i * 3 + 1) * 10 + r];
    o[20 + r] = attn_sample[(i * 3 + 1) * 10 + r];
  }
  {
    float t1[6], t2[10], xr[40];
    MATVEC_TANH(t1, 6, o, 30, d21W, d21b);
    MATVEC_TANH(t2, 10, t1, 6, d22W, d22b);
    MATVEC(xr, 40, t2, 10, d23W, d23b);
#pragma unroll
    for (int c = 0; c < 40; ++c) {
      dout[OFF_XREC + (size_t)i * 136 + 64 + c] = xr[c];
      dout[OFF_XR2 + (size_t)i * 40 + c] = xr[c];
    }
  }
  // modality 3
#pragma unroll
  for (int r = 0; r < 10; ++r) {
    o[r]      = Qb[(i * 3 + 2) * 10 + r];
    o[10 + r] = sup_attn[(i * 3 + 2) * 10 + r];
    o[20 + r] = attn_sample[(i * 3 + 2) * 10 + r];
  }
  {
    float t1[5], t2[9], xr[32];
    MATVEC_TANH(t1, 5, o, 30, d31W, d31b);
    MATVEC_TANH(t2, 9, t1, 5, d32W, d32b);
    MATVEC(xr, 32, t2, 9, d33W, d33b);
#pragma unroll
    for (int c = 0; c < 32; ++c) {
      dout[OFF_XREC + (size_t)i * 136 + 104 + c] = xr[c];
      dout[OFF_XR3 + (size_t)i * 32 + c] = xr[c];
    }
  }
}

// ---------------------------------------------------------------------------
extern "C" void kernel_launch(void* const* d_in, const int* in_sizes, int n_in,
                              void* d_out, int out_size, void* d_ws, size_t ws_size,
                              hipStream_t stream) {
  const float* X1 = (const float*)d_in[0];
  const float* X2 = (const float*)d_in[1];
  const float* X3 = (const float*)d_in[2];
  // params in setup_inputs() insertion order
  enum {
    e11_W = 0, e11_b, e12_W, e12_b, a1_W, a1_b, e21_W, e21_b, e22_W, e22_b,
    a2_W, a2_b, e31_W, e31_b, e32_W, e32_b, a3_W, a3_b, B1_W, B1_b, B2_W, B2_b,
    B3_W, B3_b, L1_W, L1_b, d11_W, d11_b, d12_W, d12_b, d13_W, d13_b, d21_W,
    d21_b, d22_W, d22_b, d23_W, d23_b, d31_W, d31_b, d32_W, d32_b, d33_W,
    d33_b, aWq, abq, aWk, abk, aWv, abv, aWf, abf, NPARAM
  };
  const float* P[NPARAM];
  for (int k = 0; k < NPARAM && 3 + k < n_in; ++k)
    P[k] = (const float*)d_in[3 + k];

  float* ws = (float*)d_ws;
  float* Qb          = ws;                    // 2048*30
  float* attn_block  = Qb + 61440;            // 2048*30
  float* out_static  = attn_block + 61440;    // 2048*30
  float* sup         = out_static + 61440;    // 2048*30
  float* sup_attn    = sup + 61440;           // 2048*30
  float* attn_sample = sup_attn + 61440;      // 2048*30
  float* e11buf      = attn_sample + 61440;   // 2048*16
  float* Lq = e11buf + 32768;                 // 6 x 6144 stats
  float* HQ = Lq + 6144;
  float* Aq = HQ + 6144;
  float* Lk = Aq + 6144;
  float* HK = Lk + 6144;
  float* Ak = HK + 6144;
  float* Vbuf = Ak + 6144;                    // 3*2048*10
  float* yp = Vbuf + 61440;                   // 3*2048*10
  float* t2buf = yp + 61440;                  // 2048*16

  float* dout = (float*)d_out;
  float* h1o = dout + OFF_H1;
  float* h2o = dout + OFF_H2;
  float* h3o = dout + OFF_H3;

  // head weight pointers (aW* shape (3,10,10); ab* shape (3,10))
  const float* Wq0 = P[aWq] + 0;   const float* bq0 = P[abq] + 0;
  const float* Wk0 = P[aWk] + 0;   const float* bk0 = P[abk] + 0;
  const float* Wv0 = P[aWv] + 0;   const float* bv0 = P[abv] + 0;
  const float* Wf0 = P[aWf] + 0;   const float* bf0 = P[abf] + 0;
  const float* Wq1 = P[aWq] + 100; const float* bq1 = P[abq] + 10;
  const float* Wk1 = P[aWk] + 100; const float* bk1 = P[abk] + 10;
  const float* Wv1 = P[aWv] + 100; const float* bv1 = P[abv] + 10;
  const float* Wf1 = P[aWf] + 100; const float* bf1 = P[abf] + 10;
  const float* Wq2 = P[aWq] + 200; const float* bq2 = P[abq] + 20;
  const float* Wk2 = P[aWk] + 200; const float* bk2 = P[abk] + 20;
  const float* Wv2 = P[aWv] + 200; const float* bv2 = P[abv] + 20;
  const float* Wf2 = P[aWf] + 200; const float* bf2 = P[abf] + 20;

  // 1) e11 via WMMA
  e11_wmma_kernel<<<NB / 16, 32, 0, stream>>>(X1, P[e11_W], P[e11_b], e11buf);
  // 2) rest of encoders -> Qb, h1/h2/h3
  encoder_rest_kernel<<<NB / 256, 256, 0, stream>>>(
      X2, X3, e11buf, P[e12_W], P[e12_b], P[a1_W], P[a1_b], P[e21_W], P[e21_b],
      P[e22_W], P[e22_b], P[a2_W], P[a2_b], P[e31_W], P[e31_b], P[e32_W],
      P[e32_b], P[a3_W], P[a3_b], P[B1_W], P[B1_b], P[B2_W], P[B2_b], P[B3_W],
      P[B3_b], Qb, h1o, h2o, h3o);
  // 3) attn_block = att(Qb, head 0)
  attn3_kernel<<<NB / 256, 256, 0, stream>>>(Qb, Wq0, bq0, Wk0, bk0, Wv0, bv0,
                                             Wf0, bf0, attn_block);
  // 4) out_static = L1(concat(attn_block, Qb))
  lin_l1_kernel<<<NB * 3 / 256, 256, 0, stream>>>(attn_block, Qb, P[L1_W],
                                                  P[L1_b], out_static);
  // 5-8) attn_sample = att(out_static^T, head 2)^T
  seqattn_stats_kernel<<<NB * 3 / 256, 256, 0, stream>>>(
      out_static, Wq2, bq2, Wk2, bk2, Wv2, bv2, Lq, HQ, Aq, Lk, HK, Ak, Vbuf);
  {
    const size_t lds_bytes = (size_t)(NB * 10 + 3 * NB) * sizeof(float); // 104KB
    seqattn_rows_kernel<<<NB * 3 / 8, 256, lds_bytes, stream>>>(
        Lq, HQ, Aq, Lk, HK, Ak, Vbuf, yp);
  }
  seq_cumsum_kernel<<<1, 32, 0, stream>>>(yp);
  seqattn_out_kernel<<<NB * 3 / 256, 256, 0, stream>>>(yp, Wf2, bf2,
                                                       attn_sample);
  // 9) sup = L1(concat(attn_sample, Qb))
  lin_l1_kernel<<<NB * 3 / 256, 256, 0, stream>>>(attn_sample, Qb, P[L1_W],
                                                  P[L1_b], sup);
  // 10) sup_attn = att(sup, head 1)
  attn3_kernel<<<NB / 256, 256, 0, stream>>>(sup, Wq1, bq1, Wk1, bk1, Wv1, bv1,
                                             Wf1, bf1, sup_attn);
  // 11) decoders (mod 2/3 full; mod 1 to t2) -> t2buf + outputs
  decoder_kernel<<<NB / 256, 256, 0, stream>>>(
      Qb, sup_attn, attn_sample, P[d11_W], P[d11_b], P[d12_W], P[d12_b],
      P[d21_W], P[d21_b], P[d22_W], P[d22_b], P[d23_W], P[d23_b], P[d31_W],
      P[d31_b], P[d32_W], P[d32_b], P[d33_W], P[d33_b], t2buf, dout);
  // 12) xr1 = t2 @ d13 via WMMA (dual write: x_recon + xr1)
  d13_wmma_kernel<<<(NB / 16) * 4, 32, 0, stream>>>(t2buf, P[d13_W], P[d13_b],
                                                    dout);
}